// ColorHistogramLoss_52733608460433
// MI455X (gfx1250) — compile-verified
//
#include <hip/hip_runtime.h>
#include <hip/hip_bf16.h>

typedef float  __attribute__((ext_vector_type(4))) vfloat4;
typedef float  __attribute__((ext_vector_type(2))) v2f;
typedef float  __attribute__((ext_vector_type(8))) v8f;

#define BINS        64
#define WPB         8            // waves per block (wave32 -> 256 threads)
#define THREADS     (WPB * 32)
#define HIST_SLOTS  (2 * 3 * BINS)   // [tensor][channel][bin] = 384

// ---------------------------------------------------------------- zero counters
__global__ void ch_hist_zero_kernel(unsigned int* g) {
    unsigned int t = threadIdx.x;
    if (t < HIST_SLOTS) g[t] = 0u;
}

// ---------------------------------------------------------------- main streaming histogram
// pred/target: (32,3,512,512) fp32. Channel plane = 512*512 = 2^18 elements
// = 2^16 float4s, so channel(c) of float4 index j is (j >> 16) % 3.
__global__ void __launch_bounds__(THREADS)
ch_hist_kernel(const vfloat4* __restrict__ p4,
               const vfloat4* __restrict__ t4,
               unsigned int*  __restrict__ ghist,
               unsigned int n4)
{
    __shared__ unsigned int lh[WPB * HIST_SLOTS];   // per-wave private histograms (12 KB)

    const unsigned int tid  = threadIdx.x;
    const unsigned int wave = tid >> 5;
    unsigned int* wh = lh + wave * HIST_SLOTS;

    for (unsigned int i = tid; i < WPB * HIST_SLOTS; i += THREADS) lh[i] = 0u;
    __syncthreads();

    const unsigned int stride = gridDim.x * THREADS;

    for (unsigned int j = blockIdx.x * THREADS + tid; j < n4; j += stride) {
        // speculative prefetch of the next grid-stride iteration (global_prefetch_b8)
        __builtin_prefetch((const void*)(p4 + j + stride), 0, 0);
        __builtin_prefetch((const void*)(t4 + j + stride), 0, 0);

        const unsigned int c = (j >> 16) % 3u;           // channel, uniform across the float4
        unsigned int* hp = wh + c * BINS;                // pred  hist for this channel
        unsigned int* ht = wh + 3 * BINS + c * BINS;     // target hist

        vfloat4 a = __builtin_nontemporal_load(p4 + j);  // streaming: NT 128-bit loads
        vfloat4 b = __builtin_nontemporal_load(t4 + j);

        #pragma unroll
        for (int k = 0; k < 4; ++k) {
            float x = a[k];
            if (x >= 0.0f && x <= 1.0f) {
                int bin = (int)(x * (float)BINS);
                atomicAdd(&hp[bin < (BINS - 1) ? bin : (BINS - 1)], 1u);  // ds_add_u32
            }
            float y = b[k];
            if (y >= 0.0f && y <= 1.0f) {
                int bin = (int)(y * (float)BINS);
                atomicAdd(&ht[bin < (BINS - 1) ? bin : (BINS - 1)], 1u);
            }
        }
    }
    __syncthreads();

    // merge the 8 per-wave histograms, one global atomic per slot per block
    for (unsigned int i = tid; i < HIST_SLOTS; i += THREADS) {
        unsigned int s = 0;
        #pragma unroll
        for (int w = 0; w < WPB; ++w) s += lh[w * HIST_SLOTS + i];
        if (s) atomicAdd(&ghist[i], s);
    }
}

// ---------------------------------------------------------------- finalize (1 wave32)
// ghist layout: [tensor(2)][channel(3)][bin(64)], tensor0 = pred.
// Sums 192 |ph - th| values exactly in fp32 via V_WMMA_F32_16X16X4_F32 with B = ones:
// every D entry is a total row-sum, so Sum(D over 256 entries) = 16 * Sum(diffs).
__global__ void __launch_bounds__(32)
ch_hist_finalize_kernel(const unsigned int* __restrict__ ghist,
                        float* __restrict__ out)
{
    __shared__ float diff[192];
    __shared__ float sums[6];      // psum[0..2], tsum[0..2]

    const int lane = threadIdx.x;  // exactly one wave32, EXEC all ones

    if (lane < 6) {                // lane = tensor*3 + channel
        const unsigned int* base = ghist + lane * BINS;
        unsigned int s = 0;
        #pragma unroll
        for (int k = 0; k < BINS; ++k) s += base[k];
        sums[lane] = (float)s;
    }
    __syncthreads();

    for (int j = lane; j < 192; j += 32) {
        int c = j >> 6, b = j & 63;
        float ps = sums[c]     + 1e-7f;
        float ts = sums[3 + c] + 1e-7f;
        float ph = (float)ghist[c * BINS + b]       / ps;
        float th = (float)ghist[192 + c * BINS + b] / ts;
        diff[j] = fabsf(ph - th);
    }
    __syncthreads();

    // A-matrix 16x4 fp32 layout: lanes 0-15 -> {K=0,K=1}, lanes 16-31 -> {K=2,K=3}
    const int M  = lane & 15;
    const int k0 = (lane < 16) ? 0 : 2;
    v2f ones; ones.x = 1.0f; ones.y = 1.0f;   // B = 4x16 all-ones (layout-independent)

    v8f acc = {};
    #pragma unroll
    for (int chunk = 0; chunk < 3; ++chunk) {          // 3 x 64 = 192 values
        v2f a;
        a.x = diff[chunk * 64 + k0 * 16 + M];
        a.y = diff[chunk * 64 + (k0 + 1) * 16 + M];
        acc = __builtin_amdgcn_wmma_f32_16x16x4_f32(
            /*neg_a=*/false, a, /*neg_b=*/false, ones,
            /*c_mod=*/(short)0, acc, /*reuse_a=*/false, /*reuse_b=*/false);
    }

    float s = acc[0] + acc[1] + acc[2] + acc[3] + acc[4] + acc[5] + acc[6] + acc[7];
    #pragma unroll
    for (int off = 16; off > 0; off >>= 1) s += __shfl_xor(s, off, 32);

    if (lane == 0)
        out[0] = s * (1.0f / 16.0f) * (1.0f / (float)BINS) * (1.0f / 3.0f);
}

// ---------------------------------------------------------------- launcher
extern "C" void kernel_launch(void* const* d_in, const int* in_sizes, int n_in,
                              void* d_out, int out_size, void* d_ws, size_t ws_size,
                              hipStream_t stream) {
    const vfloat4* pred   = (const vfloat4*)d_in[0];
    const vfloat4* target = (const vfloat4*)d_in[1];
    unsigned int*  ghist  = (unsigned int*)d_ws;      // 384 u32 counters
    float*         out    = (float*)d_out;

    const unsigned int n  = (unsigned int)in_sizes[0];   // 25,165,824
    const unsigned int n4 = n >> 2;                      // 6,291,456 float4s

    ch_hist_zero_kernel<<<1, HIST_SLOTS, 0, stream>>>(ghist);

    unsigned int blocks = (n4 + THREADS - 1) / THREADS;
    if (blocks > 4096u) blocks = 4096u;
    ch_hist_kernel<<<blocks, THREADS, 0, stream>>>(pred, target, ghist, n4);

    ch_hist_finalize_kernel<<<1, 32, 0, stream>>>(ghist, out);
}